// Parallel_calc_kernel_1949915152858
// MI455X (gfx1250) — compile-verified
//
#include <hip/hip_runtime.h>

// CDNA5 / gfx1250: fp32 Gram-matrix groups via V_WMMA_F32_16X16X4_F32,
// with double-buffered GLOBAL_LOAD_ASYNC_TO_LDS_B128 staging (ASYNCcnt).
// kernel[g] = J_row[:, Kg] @ J_col[:, Kg]^T ; group 3's second K-half scaled
// by 0.5 via an exact power-of-two in-LDS pass.

typedef float v2f __attribute__((ext_vector_type(2)));
typedef float v8f __attribute__((ext_vector_type(8)));
typedef int   v4i __attribute__((ext_vector_type(4)));

#define MAT   4096       // matrix dim and total P
#define KGRP  1024       // K per group
#define BM    128        // block tile M
#define BN    256        // block tile N
#define KB    32         // K chunk staged in LDS
#define LDSR  (KB + 4)   // padded row stride (dwords): 144B rows -> B128-aligned,
                         // and 36*l mod 64 distinct for l=0..15 -> conflict-free b64 reads
#define BUFSZ ((BM + BN) * LDSR)   // one double-buffer slot (floats)
#define NCH   (KGRP / KB)          // 32 chunks
#define OPS_PER_CHUNK 12           // async b128 issues per thread per chunk

#if __has_builtin(__builtin_amdgcn_global_load_async_to_lds_b128)
#define USE_ASYNC_LDS 1
#else
#define USE_ASYNC_LDS 0
#endif

#define GAS __attribute__((address_space(1)))
#define LAS __attribute__((address_space(3)))

template <int N>
__device__ __forceinline__ void wait_asynccnt() {
#if __has_builtin(__builtin_amdgcn_s_wait_asynccnt)
    __builtin_amdgcn_s_wait_asynccnt(N);
#else
    asm volatile("s_wait_asynccnt %0" :: "n"(N) : "memory");
#endif
}

__global__ __launch_bounds__(256)
void gram_groups_wmma_f32(const float* __restrict__ Jr,
                          const float* __restrict__ Jc,
                          float* __restrict__ out) {
    __shared__ float lds[2 * BUFSZ];

    const int tid    = threadIdx.x;
    const int lane   = tid & 31;
    const int wave   = tid >> 5;     // 8 waves
    const int wm     = wave & 1;     // 2 waves along M
    const int wn     = wave >> 1;    // 4 waves along N
    const int g      = blockIdx.z;   // parameter group 0..3
    const int m0     = blockIdx.y * BM;
    const int n0     = blockIdx.x * BN;
    const int kbase  = g * KGRP;

    // staging pattern: 8 threads x 16B per 32-float row
    const int lrow = tid >> 3;          // 0..31
    const int lcol = (tid & 7) * 4;     // 0,4,...,28

    // WMMA 16x16x4 fp32 fragment addressing:
    // lanes 0..15 read K = kk..kk+1, lanes 16..31 read K = kk+2..kk+3
    const int lane15 = lane & 15;
    const int khalf  = (lane >> 4) << 1;

    v8f acc[4][4];
#pragma unroll
    for (int i = 0; i < 4; ++i)
#pragma unroll
        for (int j = 0; j < 4; ++j)
            acc[i][j] = {};

#if USE_ASYNC_LDS
    // ---- async direct-to-LDS staging of one K-chunk into buffer `buf` ----
    auto issue_chunk = [&](int k0, int buf) {
        float* As = lds + buf * BUFSZ;
        float* Bs = As + BM * LDSR;
#pragma unroll
        for (int p = 0; p < BM / 32; ++p) {
            const int r = p * 32 + lrow;
            __builtin_amdgcn_global_load_async_to_lds_b128(
                (GAS v4i*)(Jr + (size_t)(m0 + r) * MAT + (kbase + k0 + lcol)),
                (LAS v4i*)(As + r * LDSR + lcol), 0, 0);
        }
#pragma unroll
        for (int p = 0; p < BN / 32; ++p) {
            const int r = p * 32 + lrow;
            __builtin_amdgcn_global_load_async_to_lds_b128(
                (GAS v4i*)(Jc + (size_t)(n0 + r) * MAT + (kbase + k0 + lcol)),
                (LAS v4i*)(Bs + r * LDSR + lcol), 0, 0);
        }
    };

    issue_chunk(0, 0);

    for (int i = 0; i < NCH; ++i) {
        const int cur = i & 1;
        const int k0  = i * KB;
        float* As = lds + cur * BUFSZ;
        float* Bs = As + BM * LDSR;

        // overlap: issue next chunk into the other buffer, then wait only
        // for the OPS_PER_CHUNK older (in-order) async ops of this chunk.
        if (i + 1 < NCH) {
            issue_chunk(k0 + KB, cur ^ 1);
            wait_asynccnt<OPS_PER_CHUNK>();
        } else {
            wait_asynccnt<0>();
        }
        __syncthreads();

        // prior-variance 0.5 (exact pow2) on group 3, local K >= 512:
        // async DMA can't scale in flight -> tiny in-LDS pass (16/128 chunks).
        if (g == 3 && k0 >= 512) {
            float* brow = Bs + tid * LDSR;   // one 32-float row per thread
#pragma unroll
            for (int c = 0; c < KB; c += 2) {
                v2f t = *reinterpret_cast<v2f*>(brow + c);
                t.x *= 0.5f; t.y *= 0.5f;
                *reinterpret_cast<v2f*>(brow + c) = t;
            }
            __syncthreads();
        }

        // ---- compute: 8 K-steps of 4, 16 WMMAs each ----
#pragma unroll
        for (int kk = 0; kk < KB; kk += 4) {
            v2f a[4], b[4];
#pragma unroll
            for (int ii = 0; ii < 4; ++ii)
                a[ii] = *reinterpret_cast<const v2f*>(
                    As + (wm * 64 + ii * 16 + lane15) * LDSR + kk + khalf);
#pragma unroll
            for (int jj = 0; jj < 4; ++jj)
                b[jj] = *reinterpret_cast<const v2f*>(
                    Bs + (wn * 64 + jj * 16 + lane15) * LDSR + kk + khalf);
#pragma unroll
            for (int ii = 0; ii < 4; ++ii)
#pragma unroll
                for (int jj = 0; jj < 4; ++jj)
                    acc[ii][jj] = __builtin_amdgcn_wmma_f32_16x16x4_f32(
                        false, a[ii], false, b[jj],
                        (short)0, acc[ii][jj], false, false);
        }
        __syncthreads();   // protects buf[cur] before it is re-issued at i+2
    }
#else
    // ---- fallback: synchronous float4 staging ----
    float* As = lds;
    float* Bs = lds + BM * LDSR;
    for (int k0 = 0; k0 < KGRP; k0 += KB) {
        const float scale = (g == 3 && k0 >= 512) ? 0.5f : 1.0f;
#pragma unroll
        for (int p = 0; p < BM / 32; ++p) {
            const int r = p * 32 + lrow;
            const float4 v = *reinterpret_cast<const float4*>(
                Jr + (size_t)(m0 + r) * MAT + (kbase + k0 + lcol));
            float* dst = As + r * LDSR + lcol;
            v2f lo = {v.x, v.y}; v2f hi = {v.z, v.w};
            *reinterpret_cast<v2f*>(dst)     = lo;
            *reinterpret_cast<v2f*>(dst + 2) = hi;
        }
#pragma unroll
        for (int p = 0; p < BN / 32; ++p) {
            const int r = p * 32 + lrow;
            const float4 v = *reinterpret_cast<const float4*>(
                Jc + (size_t)(n0 + r) * MAT + (kbase + k0 + lcol));
            float* dst = Bs + r * LDSR + lcol;
            v2f lo = {v.x * scale, v.y * scale}; v2f hi = {v.z * scale, v.w * scale};
            *reinterpret_cast<v2f*>(dst)     = lo;
            *reinterpret_cast<v2f*>(dst + 2) = hi;
        }
        __syncthreads();
#pragma unroll
        for (int kk = 0; kk < KB; kk += 4) {
            v2f a[4], b[4];
#pragma unroll
            for (int ii = 0; ii < 4; ++ii)
                a[ii] = *reinterpret_cast<const v2f*>(
                    As + (wm * 64 + ii * 16 + lane15) * LDSR + kk + khalf);
#pragma unroll
            for (int jj = 0; jj < 4; ++jj)
                b[jj] = *reinterpret_cast<const v2f*>(
                    Bs + (wn * 64 + jj * 16 + lane15) * LDSR + kk + khalf);
#pragma unroll
            for (int ii = 0; ii < 4; ++ii)
#pragma unroll
                for (int jj = 0; jj < 4; ++jj)
                    acc[ii][jj] = __builtin_amdgcn_wmma_f32_16x16x4_f32(
                        false, a[ii], false, b[jj],
                        (short)0, acc[ii][jj], false, false);
        }
        __syncthreads();
    }
#endif

    // ---- store: C/D layout VGPR r -> rows r (lanes 0-15) / r+8 (lanes 16-31)
    float* outg = out + (size_t)g * MAT * MAT;
    const int rhalf = (lane >> 4) * 8;
#pragma unroll
    for (int i = 0; i < 4; ++i) {
        const int mrow = m0 + wm * 64 + i * 16 + rhalf;
#pragma unroll
        for (int j = 0; j < 4; ++j) {
            const int ncol = n0 + wn * 64 + j * 16 + lane15;
#pragma unroll
            for (int r = 0; r < 8; ++r)
                outg[(size_t)(mrow + r) * MAT + ncol] = acc[i][j][r];
        }
    }
}

extern "C" void kernel_launch(void* const* d_in, const int* in_sizes, int n_in,
                              void* d_out, int out_size, void* d_ws, size_t ws_size,
                              hipStream_t stream) {
    const float* Jr = (const float*)d_in[0];   // J_row: 4096 x 4096 fp32
    const float* Jc = (const float*)d_in[1];   // J_col: 4096 x 4096 fp32
    float* out = (float*)d_out;                // [4, 4096, 4096] fp32

    dim3 grid(MAT / BN, MAT / BM, 4);          // (16, 32, 4)
    dim3 block(256);                           // 8 wave32s
    gram_groups_wmma_f32<<<grid, block, 0, stream>>>(Jr, Jc, out);
}